// Repro_11879879542435
// MI455X (gfx1250) — compile-verified
//
#include <hip/hip_runtime.h>

// Problem constants (match reference)
#define NB 1024   // batch
#define NN 21     // nodes (20 customers + depot)
#define DM 128    // d_model
#define EH 16     // head dim
#define NH 8      // heads
#define FF 512    // ffn dim

// Padded LDS strides. Even values keep every A-fragment pair 8B-aligned so
// the two K-contiguous elements merge into one ds_load_b64; strides are
// ==2 (mod 64) / ==18,34 so half-wave fragment accesses stay spread over banks.
#define HP 130    // stride for [32][128] tiles
#define QP 18     // stride for [32][16] tiles
#define SP 34     // stride for [32][32] tiles

typedef __attribute__((ext_vector_type(2))) float v2f;
typedef __attribute__((ext_vector_type(8))) float v8f;

// ---------------- WMMA helpers (V_WMMA_F32_16X16X4_F32) ----------------
// 8-arg form: (neg_a, A, neg_b, B, c_mod, C, reuse_a, reuse_b)
__device__ __forceinline__ v8f wmma4(v2f a, v2f b, v8f c) {
  return __builtin_amdgcn_wmma_f32_16x16x4_f32(false, a, false, b, (short)0, c,
                                               false, false);
}

// A fragment (16x4 f32), K contiguous in memory: lane<16 holds row m=lane,
// K={0,1}; lane>=16 holds K={2,3}. Single 64-bit load (8B aligned by stride).
__device__ __forceinline__ v2f frag_a(const float* p, int rs, int lane) {
  int m = lane & 15;
  int kb = (lane >> 4) << 1;
  return *(const v2f*)(p + m * rs + kb);
}
// B fragment (4x16 f32), generic strides: element (k,n) at p[k*rs + n*cs]
__device__ __forceinline__ v2f frag_b(const float* p, int rs, int cs, int lane) {
  int n = lane & 15;
  int kb = (lane >> 4) << 1;
  v2f f;
  f.x = p[kb * rs + n * cs];
  f.y = p[(kb + 1) * rs + n * cs];
  return f;
}
// B fragment with rs==1 (transposed source, K contiguous): one 64-bit load.
__device__ __forceinline__ v2f frag_bt(const float* p, int cs, int lane) {
  int n = lane & 15;
  int kb = (lane >> 4) << 1;
  return *(const v2f*)(p + n * cs + kb);
}
// C/D 16x16 f32 tile: VGPR r -> row (lane<16 ? r : r+8), col = lane&15
__device__ __forceinline__ void store_c(float* p, int rs, v8f c, int lane) {
  int n = lane & 15;
  int mo = (lane >> 4) << 3;
#pragma unroll
  for (int r = 0; r < 8; ++r) p[(mo + r) * rs + n] = c[r];
}

// One [32x128] @ [128x16] projection: sh_h (LDS) x W (global, row-major DxE)
// -> dst (LDS, stride QP). Direct args keep pointer provenance visible so the
// backend emits global_load (weights) + ds_load/ds_store (LDS), not flat.
__device__ __forceinline__ void proj_128x16(const float* sh_h,
                                            const float* __restrict__ W,
                                            float* dst, int lane) {
#pragma unroll
  for (int mt = 0; mt < 2; ++mt) {
    v8f c = {};
    for (int k0 = 0; k0 < DM; k0 += 4) {
      v2f a = frag_a(sh_h + mt * 16 * HP + k0, HP, lane);
      v2f bb = frag_b(W + k0 * EH, EH, 1, lane);
      c = wmma4(a, bb, c);
    }
    store_c(dst + mt * 16 * QP, QP, c, lane);
  }
}

// ---------------- embedding ----------------
__global__ void embed_kernel(const float* __restrict__ loc,
                             const float* __restrict__ dem,
                             const float* __restrict__ dep,
                             const float* __restrict__ dw,
                             const float* __restrict__ db,
                             const float* __restrict__ nw,
                             const float* __restrict__ nb,
                             float* __restrict__ h) {
  int idx = blockIdx.x * blockDim.x + threadIdx.x;
  if (idx >= NB * NN * DM) return;
  int d = idx & (DM - 1);
  int rn = idx / DM;
  int b = rn / NN;
  int r = rn - b * NN;
  float v;
  if (r == 0) {
    v = dep[b * 2 + 0] * dw[d] + dep[b * 2 + 1] * dw[DM + d] + db[d];
  } else {
    int n = r - 1;
    float x = loc[(b * 20 + n) * 2 + 0];
    float y = loc[(b * 20 + n) * 2 + 1];
    float q = dem[b * 20 + n];
    v = x * nw[d] + y * nw[DM + d] + q * nw[2 * DM + d] + nb[d];
  }
  h[idx] = v;
}

// ---------------- attention (one block per graph, one wave per head) ----------------
__global__ void __launch_bounds__(256) attn_kernel(
    float* __restrict__ h, const float* __restrict__ Wq,
    const float* __restrict__ Wk, const float* __restrict__ Wv,
    const float* __restrict__ Wo, float* __restrict__ stats) {
  extern __shared__ float smem[];
  float* sh_h = smem;                    // 32*HP
  float* sh_q = sh_h + 32 * HP;          // NH*32*QP (becomes o-matrix later)
  float* sh_k = sh_q + NH * 32 * QP;     // NH*32*QP
  float* sh_v = sh_k + NH * 32 * QP;     // NH*32*QP
  float* sh_s = sh_v + NH * 32 * QP;     // NH*32*SP

  const int tid = threadIdx.x;
  const int lane = tid & 31;
  const int wv = tid >> 5;  // wave id == head id
  const int b = blockIdx.x;
  float* gh = h + b * NN * DM;

  // stage h tile to LDS, zero-pad rows 21..31
  for (int i = tid; i < 32 * DM; i += 256) {
    int r = i >> 7, d = i & 127;
    sh_h[r * HP + d] = (r < NN) ? gh[r * DM + d] : 0.f;
  }
  __syncthreads();

  float* qh = sh_q + wv * 32 * QP;
  float* kh = sh_k + wv * 32 * QP;
  float* vh = sh_v + wv * 32 * QP;
  float* shs = sh_s + wv * 32 * SP;

  // ---- QKV: [32x128] @ [128x16] per head ----
  proj_128x16(sh_h, Wq + wv * DM * EH, qh, lane);
  proj_128x16(sh_h, Wk + wv * DM * EH, kh, lane);
  proj_128x16(sh_h, Wv + wv * DM * EH, vh, lane);

  // ---- scores: s = q @ k^T (k^T fragment is K-contiguous -> ds_load_b64) ----
#pragma unroll
  for (int mt = 0; mt < 2; ++mt)
#pragma unroll
    for (int ntile = 0; ntile < 2; ++ntile) {
      v8f c = {};
#pragma unroll
      for (int k0 = 0; k0 < EH; k0 += 4) {
        v2f a = frag_a(qh + mt * 16 * QP + k0, QP, lane);
        v2f bb = frag_bt(kh + ntile * 16 * QP + k0, QP, lane);
        c = wmma4(a, bb, c);
      }
      store_c(shs + mt * 16 * SP + ntile * 16, SP, c, lane);
    }

  // ---- masked softmax over 21 columns (lane = row, scale 0.25 folded in) ----
  {
    int i = lane;
    float mx = -1e30f;
    for (int j = 0; j < NN; ++j) mx = fmaxf(mx, 0.25f * shs[i * SP + j]);
    float sum = 0.f;
    for (int j = 0; j < NN; ++j) sum += __expf(0.25f * shs[i * SP + j] - mx);
    float inv = 1.f / sum;
    for (int j = 0; j < NN; ++j)
      shs[i * SP + j] = __expf(0.25f * shs[i * SP + j] - mx) * inv;
    for (int j = NN; j < 32; ++j) shs[i * SP + j] = 0.f;  // K-padding
  }

  // ---- o = a @ v  (K padded to 32; padded cols of a are zero) ----
#pragma unroll
  for (int mt = 0; mt < 2; ++mt) {
    v8f c = {};
#pragma unroll
    for (int k0 = 0; k0 < 32; k0 += 4) {
      v2f a = frag_a(shs + mt * 16 * SP + k0, SP, lane);
      v2f bb = frag_b(vh + k0 * QP, QP, 1, lane);
      c = wmma4(a, bb, c);
    }
    store_c(qh + mt * 16 * QP, QP, c, lane);  // o overwrites q region
  }
  __syncthreads();

  // ---- out-proj: concat_h(o)[32x128] @ stacked Wo[128x128]; wave owns col tile ----
  const int ntile = wv;
#pragma unroll
  for (int mt = 0; mt < 2; ++mt) {
    v8f c = {};
    for (int kh2 = 0; kh2 < NH; ++kh2) {
      const float* Oh = sh_q + kh2 * 32 * QP + mt * 16 * QP;
      const float* Wh = Wo + kh2 * EH * DM + ntile * 16;
#pragma unroll
      for (int k0 = 0; k0 < EH; k0 += 4) {
        v2f a = frag_a(Oh + k0, QP, lane);
        v2f bb = frag_b(Wh + k0 * DM, DM, 1, lane);
        c = wmma4(a, bb, c);
      }
    }
    // residual add, in place (each element owned by exactly one lane)
    int n = lane & 15, mo = (lane >> 4) << 3;
#pragma unroll
    for (int r = 0; r < 8; ++r)
      sh_h[(mt * 16 + mo + r) * HP + ntile * 16 + n] += c[r];
  }
  __syncthreads();

  // ---- writeback + BN partial stats (sum, sumsq per channel) ----
  for (int i = tid; i < NN * DM; i += 256) {
    int r = i >> 7, d = i & 127;
    gh[i] = sh_h[r * HP + d];
  }
  if (tid < DM) {
    float s1 = 0.f, s2 = 0.f;
    for (int r = 0; r < NN; ++r) {
      float v = sh_h[r * HP + tid];
      s1 += v;
      s2 += v * v;
    }
    atomicAdd(&stats[tid], s1);
    atomicAdd(&stats[DM + tid], s2);
  }
}

// ---------------- FFN (hidden kept in LDS, 4 chunks of 128) ----------------
__global__ void __launch_bounds__(256) ffn_kernel(
    float* __restrict__ h, const float* __restrict__ w1,
    const float* __restrict__ b1, const float* __restrict__ w2,
    const float* __restrict__ b2, float* __restrict__ stats) {
  __shared__ float sh_h[32 * HP];
  __shared__ float sh_t[32 * HP];
  const int tid = threadIdx.x;
  const int lane = tid & 31;
  const int wv = tid >> 5;
  const int b = blockIdx.x;
  float* gh = h + b * NN * DM;

  for (int i = tid; i < 32 * DM; i += 256) {
    int r = i >> 7, d = i & 127;
    sh_h[r * HP + d] = (r < NN) ? gh[r * DM + d] : 0.f;
  }
  __syncthreads();

  const int ntile = wv;
  v8f acc[2];
  {
    float bv = b2[ntile * 16 + (lane & 15)];  // fold output bias into C init
#pragma unroll
    for (int r = 0; r < 8; ++r) {
      acc[0][r] = bv;
      acc[1][r] = bv;
    }
  }

  for (int ch = 0; ch < FF / DM; ++ch) {
    // hidden chunk = relu(h @ w1[:, ch*128 : (ch+1)*128] + b1)
#pragma unroll
    for (int mt = 0; mt < 2; ++mt) {
      v8f c = {};
      for (int k0 = 0; k0 < DM; k0 += 4) {
        v2f a = frag_a(sh_h + mt * 16 * HP + k0, HP, lane);
        v2f bb = frag_b(w1 + k0 * FF + ch * DM + ntile * 16, FF, 1, lane);
        c = wmma4(a, bb, c);
      }
      int n = lane & 15, mo = (lane >> 4) << 3;
      float bias = b1[ch * DM + ntile * 16 + n];
#pragma unroll
      for (int r = 0; r < 8; ++r)
        sh_t[(mt * 16 + mo + r) * HP + ntile * 16 + n] =
            fmaxf(c[r] + bias, 0.f);
    }
    __syncthreads();
    // acc += hidden @ w2[ch*128 : (ch+1)*128, :]
#pragma unroll
    for (int mt = 0; mt < 2; ++mt) {
      for (int k0 = 0; k0 < DM; k0 += 4) {
        v2f a = frag_a(sh_t + mt * 16 * HP + k0, HP, lane);
        v2f bb = frag_b(w2 + (ch * DM + k0) * DM + ntile * 16, DM, 1, lane);
        acc[mt] = wmma4(a, bb, acc[mt]);
      }
    }
    __syncthreads();
  }

  // residual add in place
#pragma unroll
  for (int mt = 0; mt < 2; ++mt) {
    int n = lane & 15, mo = (lane >> 4) << 3;
#pragma unroll
    for (int r = 0; r < 8; ++r)
      sh_h[(mt * 16 + mo + r) * HP + ntile * 16 + n] += acc[mt][r];
  }
  __syncthreads();

  for (int i = tid; i < NN * DM; i += 256) {
    int r = i >> 7, d = i & 127;
    gh[i] = sh_h[r * HP + d];
  }
  if (tid < DM) {
    float s1 = 0.f, s2 = 0.f;
    for (int r = 0; r < NN; ++r) {
      float v = sh_h[r * HP + tid];
      s1 += v;
      s2 += v * v;
    }
    atomicAdd(&stats[tid], s1);
    atomicAdd(&stats[DM + tid], s2);
  }
}

// ---------------- BatchNorm glue ----------------
__global__ void zero_stats(float* stats) {
  if (threadIdx.x < 2 * DM) stats[threadIdx.x] = 0.f;
}

__global__ void bn_prep(float* stats, const float* __restrict__ g,
                        const float* __restrict__ bb, float inv_n) {
  int d = threadIdx.x;
  if (d < DM) {
    float m = stats[d] * inv_n;
    float var = stats[DM + d] * inv_n - m * m;  // biased variance
    float sc = g[d] * rsqrtf(var + 1e-5f);
    stats[d] = sc;                   // scale
    stats[DM + d] = bb[d] - m * sc;  // shift
  }
}

// float4-vectorized elementwise normalize: y = y*scale[d] + shift[d]
__global__ void bn_apply(float4* __restrict__ y, const float* __restrict__ stats,
                         int total4) {
  int idx = blockIdx.x * blockDim.x + threadIdx.x;
  if (idx < total4) {
    int d4 = (idx & (DM / 4 - 1)) << 2;  // first channel of this float4
    float4 v = y[idx];
    v.x = v.x * stats[d4 + 0] + stats[DM + d4 + 0];
    v.y = v.y * stats[d4 + 1] + stats[DM + d4 + 1];
    v.z = v.z * stats[d4 + 2] + stats[DM + d4 + 2];
    v.w = v.w * stats[d4 + 3] + stats[DM + d4 + 3];
    y[idx] = v;
  }
}

// ---------------- host ----------------
extern "C" void kernel_launch(void* const* d_in, const int* in_sizes, int n_in,
                              void* d_out, int out_size, void* d_ws,
                              size_t ws_size, hipStream_t stream) {
  const float* loc = (const float*)d_in[0];
  const float* dem = (const float*)d_in[1];
  const float* dep = (const float*)d_in[2];
  const float* dw = (const float*)d_in[3];
  const float* db = (const float*)d_in[4];
  const float* nw = (const float*)d_in[5];
  const float* nb = (const float*)d_in[6];
  const float* Wq = (const float*)d_in[7];
  const float* Wk = (const float*)d_in[8];
  const float* Wv = (const float*)d_in[9];
  const float* Wo = (const float*)d_in[10];
  const float* g1 = (const float*)d_in[11];
  const float* bb1 = (const float*)d_in[12];
  const float* w1 = (const float*)d_in[13];
  const float* fb1 = (const float*)d_in[14];
  const float* w2 = (const float*)d_in[15];
  const float* fb2 = (const float*)d_in[16];
  const float* g2 = (const float*)d_in[17];
  const float* bb2 = (const float*)d_in[18];

  float* h = (float*)d_out;     // evolve h in place in the output buffer
  float* stats = (float*)d_ws;  // 256 floats of scratch
  const int total = NB * NN * DM;
  const int total4 = total / 4;

  const size_t attn_lds =
      (size_t)(32 * HP + 3 * NH * 32 * QP + NH * 32 * SP) * sizeof(float);
  hipFuncSetAttribute((const void*)attn_kernel,
                      hipFuncAttributeMaxDynamicSharedMemorySize,
                      (int)attn_lds);

  embed_kernel<<<(total + 255) / 256, 256, 0, stream>>>(loc, dem, dep, dw, db,
                                                        nw, nb, h);

  const float inv_n = 1.f / (float)(NB * NN);
  for (int l = 0; l < 3; ++l) {
    zero_stats<<<1, 256, 0, stream>>>(stats);
    attn_kernel<<<NB, 256, attn_lds, stream>>>(
        h, Wq + l * NH * DM * EH, Wk + l * NH * DM * EH, Wv + l * NH * DM * EH,
        Wo + l * NH * EH * DM, stats);
    bn_prep<<<1, 128, 0, stream>>>(stats, g1 + l * DM, bb1 + l * DM, inv_n);
    bn_apply<<<(total4 + 255) / 256, 256, 0, stream>>>((float4*)h, stats,
                                                       total4);

    zero_stats<<<1, 256, 0, stream>>>(stats);
    ffn_kernel<<<NB, 256, 0, stream>>>(h, w1 + l * DM * FF, fb1 + l * FF,
                                       w2 + l * FF * DM, fb2 + l * DM, stats);
    bn_prep<<<1, 128, 0, stream>>>(stats, g2 + l * DM, bb2 + l * DM, inv_n);
    bn_apply<<<(total4 + 255) / 256, 256, 0, stream>>>((float4*)h, stats,
                                                       total4);
  }
  (void)in_sizes;
  (void)n_in;
  (void)out_size;
  (void)ws_size;
}